// StridedMaxPoolLayer_64665027608656
// MI455X (gfx1250) — compile-verified
//
#include <hip/hip_runtime.h>
#include <float.h>
#include <stdint.h>

#define AS1 __attribute__((address_space(1)))
#define AS3 __attribute__((address_space(3)))

typedef int   v4i __attribute__((ext_vector_type(4)));
typedef float v4f __attribute__((ext_vector_type(4)));
typedef AS1 v4i* gv4p;   // global (addrspace 1) pointer to 128-bit vector
typedef AS3 v4i* lv4p;   // LDS (addrspace 3) pointer to 128-bit vector

#if defined(__has_builtin)
#if __has_builtin(__builtin_amdgcn_global_load_async_to_lds_b128)
#define HAVE_ASYNC_B128 1
#endif
#if __has_builtin(__builtin_amdgcn_s_wait_asynccnt)
#define HAVE_WAIT_ASYNC 1
#endif
#endif

// ---------------- constants for the 29-bit key space ----------------
// key = batch[8b] << 21 | (x>>1)[7b] << 14 | (y>>1)[7b] << 7 | (z>>1)[7b]
// numeric order of key == lexicographic order of down-sampled [b,x,y,z] rows,
// which is exactly jnp.unique's sort order.
#define NUM_WORDS (1u << 24)       // 2^29 bits / 32
#define WORDS_PER_BLOCK 4096u      // 256 threads * 16 words
#define NBLK 4096u                 // NUM_WORDS / WORDS_PER_BLOCK

__device__ __forceinline__ unsigned pack_key(int4 c) {
    return ((unsigned)c.x << 21) | ((unsigned)(c.y >> 1) << 14) |
           ((unsigned)(c.z >> 1) << 7) | (unsigned)(c.w >> 1);
}

// Exact IEEE-ordered float max using only native single-instruction atomics.
// Non-negative bit patterns (sign==0): int ordering == float ordering -> atomicMax(i32).
// Sign-set patterns (incl. -0.0):      uint ordering is reversed      -> atomicMin(u32).
// Mixed-sign correctness: any sign==0 update beats any sign==1 content under
// int-max, and any sign==1 update loses to sign==0 content under uint-min.
__device__ __forceinline__ void atomicMaxF(float* addr, float v) {
    unsigned b = __float_as_uint(v);
    if ((b >> 31) == 0u) atomicMax((int*)addr, (int)b);
    else                 atomicMin((unsigned*)addr, b);
}

// ---------------- 1) build presence bitmap ----------------
__global__ void k_build_bitmap(const int4* __restrict__ coords,
                               unsigned* __restrict__ bitmap, int N) {
    int i = blockIdx.x * blockDim.x + threadIdx.x;
    if (i >= N) return;
    unsigned key = pack_key(coords[i]);
    atomicOr(&bitmap[key >> 5], 1u << (key & 31u));
}

// ---------------- 2) per-block popcount sums (async-LDS double buffered) ----
__global__ void k_popcount_blocksums(const uint4* __restrict__ bm4,
                                     unsigned* __restrict__ blockSums) {
    __shared__ unsigned red[256];
    const unsigned t = threadIdx.x;
    const size_t base = (size_t)blockIdx.x * 1024u;  // uint4 index
    unsigned sum = 0;
#if defined(HAVE_ASYNC_B128)
    __shared__ uint4 stage[2][256];
    // issue chunk 0: each lane DMAs its own 16B slot into LDS
    __builtin_amdgcn_global_load_async_to_lds_b128(
        (gv4p)(uintptr_t)(bm4 + base + t),
        (lv4p)(unsigned)(uintptr_t)&stage[0][t], 0, 0);
    for (int k = 0; k < 4; ++k) {
        if (k < 3) {
            __builtin_amdgcn_global_load_async_to_lds_b128(
                (gv4p)(uintptr_t)(bm4 + base + (size_t)(k + 1) * 256u + t),
                (lv4p)(unsigned)(uintptr_t)&stage[(k + 1) & 1][t], 0, 0);
#if defined(HAVE_WAIT_ASYNC)
            __builtin_amdgcn_s_wait_asynccnt(1);
#else
            asm volatile("s_wait_asynccnt 1" ::: "memory");
#endif
        } else {
#if defined(HAVE_WAIT_ASYNC)
            __builtin_amdgcn_s_wait_asynccnt(0);
#else
            asm volatile("s_wait_asynccnt 0" ::: "memory");
#endif
        }
        uint4 v = stage[k & 1][t];  // lane reads only its own slot -> no barrier
        sum += __popc(v.x) + __popc(v.y) + __popc(v.z) + __popc(v.w);
    }
#else
    for (int k = 0; k < 4; ++k) {
        uint4 v = bm4[base + (size_t)k * 256u + t];
        sum += __popc(v.x) + __popc(v.y) + __popc(v.z) + __popc(v.w);
    }
#endif
    red[t] = sum;
    __syncthreads();
    for (unsigned s = 128; s > 0; s >>= 1) {
        if (t < s) red[t] += red[t + s];
        __syncthreads();
    }
    if (t == 0) blockSums[blockIdx.x] = red[0];
}

// ---------------- 3) exclusive scan of 4096 block sums (single block) -------
__global__ void k_scan_blocksums(unsigned* __restrict__ blockSums,
                                 unsigned* __restrict__ total) {
    __shared__ unsigned sh[256];
    const unsigned t = threadIdx.x;
    unsigned v[16], s = 0;
#pragma unroll
    for (int k = 0; k < 16; ++k) { v[k] = blockSums[t * 16 + k]; s += v[k]; }
    sh[t] = s;
    __syncthreads();
    for (unsigned off = 1; off < 256; off <<= 1) {
        unsigned x = (t >= off) ? sh[t - off] : 0u;
        __syncthreads();
        sh[t] += x;
        __syncthreads();
    }
    unsigned run = sh[t] - s;  // exclusive prefix of this thread's chunk
    if (t == 255) *total = sh[255];
#pragma unroll
    for (int k = 0; k < 16; ++k) { unsigned pv = v[k]; blockSums[t * 16 + k] = run; run += pv; }
}

// ---------------- 4) per-word exclusive prefixes ----------------------------
__global__ void k_word_prefix(const unsigned* __restrict__ bitmap,
                              const unsigned* __restrict__ blockBase,
                              unsigned* __restrict__ wordPrefix) {
    __shared__ unsigned sh[256];
    const unsigned t = threadIdx.x;
    const size_t wb = (size_t)blockIdx.x * WORDS_PER_BLOCK + (size_t)t * 16u;
    const uint4* p = (const uint4*)(bitmap + wb);
    uint4 a = p[0], b = p[1], c = p[2], d = p[3];
    unsigned w[16] = {a.x, a.y, a.z, a.w, b.x, b.y, b.z, b.w,
                      c.x, c.y, c.z, c.w, d.x, d.y, d.z, d.w};
    unsigned s = 0;
#pragma unroll
    for (int k = 0; k < 16; ++k) s += __popc(w[k]);
    sh[t] = s;
    __syncthreads();
    for (unsigned off = 1; off < 256; off <<= 1) {
        unsigned x = (t >= off) ? sh[t - off] : 0u;
        __syncthreads();
        sh[t] += x;
        __syncthreads();
    }
    unsigned run = blockBase[blockIdx.x] + sh[t] - s;
#pragma unroll
    for (int k = 0; k < 16; ++k) { wordPrefix[wb + k] = run; run += __popc(w[k]); }
}

// ------- 5) rank -> inv[i] (== jnp.unique inverse) + segment populations ----
__global__ void k_rank(const int4* __restrict__ coords,
                       const unsigned* __restrict__ bitmap,
                       const unsigned* __restrict__ wordPrefix,
                       unsigned* __restrict__ inv,
                       unsigned* __restrict__ elemCnt, int N) {
    int i = blockIdx.x * blockDim.x + threadIdx.x;
    if (i >= N) return;
    unsigned key = pack_key(coords[i]);
    unsigned w = key >> 5, b = key & 31u;
    unsigned r = wordPrefix[w] + __popc(bitmap[w] & ((1u << b) - 1u));
    inv[i] = r;
    atomicAdd(&elemCnt[r], 1u);
}

// -------- 6) init only multi-population rows to -FLT_MAX (~0.1% of rows) ----
__global__ void k_init_multi(const unsigned* __restrict__ elemCnt,
                             float* __restrict__ outf, int N, int C) {
    int j = blockIdx.x * blockDim.x + threadIdx.x;
    if (j >= N) return;
    if (elemCnt[j] > 1u) {
        for (int c = 0; c < C; ++c) outf[(size_t)j * C + c] = -FLT_MAX;
    }
}

// ---------------- 7) segment max: plain b128 store for singletons, ----------
// ----------------    native i32/u32 max/min atomics for multi rows ----------
__global__ void k_scatter_max(const v4f* __restrict__ feats4,
                              const unsigned* __restrict__ inv,
                              const unsigned* __restrict__ elemCnt,
                              float* __restrict__ outf, int N, int C4) {
    unsigned q = blockIdx.x * blockDim.x + threadIdx.x;
    if (q >= (unsigned)N * (unsigned)C4) return;
    unsigned i = q / (unsigned)C4;
    unsigned c4 = q - i * (unsigned)C4;
    unsigned j = inv[i];
    v4f v = __builtin_nontemporal_load(feats4 + q);  // read-once stream
    float* dst = outf + (size_t)j * (size_t)(C4 * 4) + (size_t)c4 * 4u;
    if (elemCnt[j] == 1u) {
        __builtin_nontemporal_store(v, (v4f*)dst);   // write-once stream
    } else {
        atomicMaxF(dst + 0, v.x);
        atomicMaxF(dst + 1, v.y);
        atomicMaxF(dst + 2, v.z);
        atomicMaxF(dst + 3, v.w);
    }
}

// ---------------- 8) centroid sums ----------------
__global__ void k_scatter_centroid(const float* __restrict__ points,
                                   const float* __restrict__ count,
                                   const unsigned* __restrict__ inv,
                                   float* __restrict__ outp,
                                   float* __restrict__ outc, int N) {
    int i = blockIdx.x * blockDim.x + threadIdx.x;
    if (i >= N) return;
    unsigned j = inv[i];
    float cnt = count[i];
    atomicAdd(&outp[(size_t)j * 3 + 0], cnt * points[(size_t)i * 3 + 0]);
    atomicAdd(&outp[(size_t)j * 3 + 1], cnt * points[(size_t)i * 3 + 1]);
    atomicAdd(&outp[(size_t)j * 3 + 2], cnt * points[(size_t)i * 3 + 2]);
    atomicAdd(&outc[j], cnt);
}

// ---------------- 9) finalize: divide centroids, zero padding rows ----------
__global__ void k_finalize(float* __restrict__ outf, float* __restrict__ outp,
                           float* __restrict__ outc,
                           const unsigned* __restrict__ total, int N, int C) {
    int j = blockIdx.x * blockDim.x + threadIdx.x;
    if (j >= N) return;
    unsigned U = *total;
    if (j < (int)U) {
        float cnt = outc[j];
        if (cnt > 0.f) {
            outp[(size_t)j * 3 + 0] /= cnt;
            outp[(size_t)j * 3 + 1] /= cnt;
            outp[(size_t)j * 3 + 2] /= cnt;
        } else {  // valid==false: feats row -> 0 (points = sums/1.0 already)
            for (int c = 0; c < C; ++c) outf[(size_t)j * C + c] = 0.f;
        }
    } else {  // jnp.unique padding segment: empty row -> all outputs 0
        for (int c = 0; c < C; ++c) outf[(size_t)j * C + c] = 0.f;
    }
}

extern "C" void kernel_launch(void* const* d_in, const int* in_sizes, int n_in,
                              void* d_out, int out_size, void* d_ws, size_t ws_size,
                              hipStream_t stream) {
    const int N = in_sizes[0] / 4;       // coords (N,4)
    const int C = in_sizes[1] / N;       // feats (N,C) ; C == 96
    const int C4 = C / 4;

    const int4*  coords = (const int4*)d_in[0];
    const float* feats  = (const float*)d_in[1];
    const float* points = (const float*)d_in[2];
    const float* count  = (const float*)d_in[3];

    float* outf = (float*)d_out;                      // N*C
    float* outp = outf + (size_t)N * (size_t)C;       // N*3
    float* outc = outp + (size_t)N * 3;               // N

    // workspace layout
    unsigned* bitmap     = (unsigned*)d_ws;           // 64 MB
    unsigned* wordPrefix = bitmap + NUM_WORDS;        // 64 MB
    unsigned* blockSums  = wordPrefix + NUM_WORDS;    // 16 KB
    unsigned* total      = blockSums + NBLK;
    unsigned* inv        = total + 64;                // N * 4B
    unsigned* elemCnt    = inv + (size_t)N;           // N * 4B

    (void)hipMemsetAsync(bitmap, 0, (size_t)NUM_WORDS * 4u, stream);
    (void)hipMemsetAsync(outp, 0, (size_t)N * 4u * sizeof(float), stream);  // points+cnt
    (void)hipMemsetAsync(elemCnt, 0, (size_t)N * sizeof(unsigned), stream);

    k_build_bitmap<<<(N + 255) / 256, 256, 0, stream>>>(coords, bitmap, N);
    k_popcount_blocksums<<<NBLK, 256, 0, stream>>>((const uint4*)bitmap, blockSums);
    k_scan_blocksums<<<1, 256, 0, stream>>>(blockSums, total);
    k_word_prefix<<<NBLK, 256, 0, stream>>>(bitmap, blockSums, wordPrefix);
    k_rank<<<(N + 255) / 256, 256, 0, stream>>>(coords, bitmap, wordPrefix, inv,
                                                elemCnt, N);
    k_init_multi<<<(N + 255) / 256, 256, 0, stream>>>(elemCnt, outf, N, C);

    unsigned nq = (unsigned)((size_t)N * (size_t)C4);
    k_scatter_max<<<(nq + 255) / 256, 256, 0, stream>>>((const v4f*)feats, inv,
                                                        elemCnt, outf, N, C4);
    k_scatter_centroid<<<(N + 255) / 256, 256, 0, stream>>>(points, count, inv,
                                                            outp, outc, N);
    k_finalize<<<(N + 255) / 256, 256, 0, stream>>>(outf, outp, outc, total, N, C);
}